// RPN_28922309771852
// MI455X (gfx1250) — compile-verified
//
#include <hip/hip_runtime.h>
#include <math.h>

// ---- problem constants (from reference) ----
#define PER        12          // PER_LOC_BIN_NUM
#define NHEAD      12
#define NCOL       76          // C = 4*12 + 1 + 2*12 + 3
#define ROWS_PB    128         // rows per block
#define THREADS    128
#define NVEC_PB    (ROWS_PB * (NCOL / 4))   // float4s of pred per block (2432)
#define NROI_PB    (ROWS_PB * 7)            // floats of roi/out per block (896)

#define TWO_PI_F   6.28318530717958647692f
#define PI_F       3.14159265358979323846f
#define APC_F      (TWO_PI_F / 12.0f)      // angle per class

// exact pointee types per the compiler diagnostic:
//   b128: 'int __attribute__((vector_size(16)))' ; b32: 'int'
typedef int vi4_t __attribute__((vector_size(16)));
typedef __attribute__((address_space(1))) vi4_t* gv4p_t;
typedef __attribute__((address_space(3))) vi4_t* lv4p_t;
typedef __attribute__((address_space(1))) int*   g32p_t;
typedef __attribute__((address_space(3))) int*   l32p_t;

// ---------- CDNA5 async global -> LDS copy helpers ----------
__device__ __forceinline__ void async_g2l_b128(const float* g, float* l) {
#if __has_builtin(__builtin_amdgcn_global_load_async_to_lds_b128)
  __builtin_amdgcn_global_load_async_to_lds_b128((gv4p_t)g, (lv4p_t)l, 0, 0);
#else
  *(float4*)l = *(const float4*)g;
#endif
}

__device__ __forceinline__ void async_g2l_b32(const float* g, float* l) {
#if __has_builtin(__builtin_amdgcn_global_load_async_to_lds_b32)
  __builtin_amdgcn_global_load_async_to_lds_b32((g32p_t)g, (l32p_t)l, 0, 0);
#else
  *l = *g;
#endif
}

__device__ __forceinline__ void wait_async_lds() {
#if __has_builtin(__builtin_amdgcn_s_wait_asynccnt)
  __builtin_amdgcn_s_wait_asynccnt(0);
#else
  asm volatile("s_wait_asynccnt 0" ::: "memory");
#endif
}

// -------------------------------------------------------------
__global__ __launch_bounds__(THREADS)
void rpn_decode_kernel(const float* __restrict__ roi,
                       const float* __restrict__ pred,
                       const float* __restrict__ anchor,
                       float* __restrict__ out, int n) {
  __shared__ __align__(16) float sm_pred[ROWS_PB * NCOL];  // 38 912 B
  __shared__ __align__(16) float sm_roi [NROI_PB];         //  3 584 B
  __shared__ __align__(16) float sm_out [NROI_PB];         //  3 584 B

  const int tid  = threadIdx.x;
  const int base = blockIdx.x * ROWS_PB;
  int rows = n - base;
  if (rows > ROWS_PB) rows = ROWS_PB;

  const float* pbase = pred + (size_t)base * NCOL;
  const float* rbase = roi  + (size_t)base * 7;

  if (rows == ROWS_PB) {
    // ---- common case: full tile, straight-line unguarded async copies ----
#pragma unroll
    for (int i = 0; i < NVEC_PB / THREADS; ++i) {
      const int idx = i * THREADS + tid;
      async_g2l_b128(pbase + idx * 4, &sm_pred[idx * 4]);
    }
#pragma unroll
    for (int i = 0; i < NROI_PB / THREADS; ++i) {
      const int idx = i * THREADS + tid;
      async_g2l_b32(rbase + idx, &sm_roi[idx]);
    }
  } else {
    // ---- tail block: per-element guards ----
    const int nvec = rows * (NCOL / 4);
#pragma unroll
    for (int i = 0; i < NVEC_PB / THREADS; ++i) {
      const int idx = i * THREADS + tid;
      if (idx < nvec) async_g2l_b128(pbase + idx * 4, &sm_pred[idx * 4]);
    }
    const int nsc = rows * 7;
#pragma unroll
    for (int i = 0; i < NROI_PB / THREADS; ++i) {
      const int idx = i * THREADS + tid;
      if (idx < nsc) async_g2l_b32(rbase + idx, &sm_roi[idx]);
    }
  }

  // anchor is tiny & uniform -> scalar cached loads
  const float a0 = anchor[0], a1 = anchor[1], a2 = anchor[2];

  wait_async_lds();       // my wave's async LDS writes complete
  __syncthreads();        // everyone's staged data visible

  // ---- per-row decode (lane tid owns row base+tid) ----
  if (tid < rows) {
    const float* p = &sm_pred[tid * NCOL];
    const float* r = &sm_roi[tid * 7];

    // first-occurrence argmax over x / z location bins
    int   xb = 0;  float xm = p[0];
    int   zb = 0;  float zm = p[PER];
#pragma unroll
    for (int i = 1; i < PER; ++i) {
      float vx = p[i];        if (vx > xm) { xm = vx; xb = i; }
      float vz = p[PER + i];  if (vz > zm) { zm = vz; zb = i; }
    }
    float pos_x = (float)xb * 0.5f + 0.25f - 3.0f + p[2 * PER + xb] * 0.5f;
    float pos_z = (float)zb * 0.5f + 0.25f - 3.0f + p[3 * PER + zb] * 0.5f;

    float pos_y = r[1] + p[4 * PER];          // col 48

    // heading bin argmax (cols 49..60), residual (cols 61..72)
    int   rb = 0; float rm = p[49];
#pragma unroll
    for (int i = 1; i < NHEAD; ++i) {
      float v = p[49 + i];    if (v > rm) { rm = v; rb = i; }
    }
    float ry = (float)rb * APC_F + p[61 + rb] * (APC_F * 0.5f);
    ry = ry - TWO_PI_F * floorf(ry * (1.0f / TWO_PI_F));   // python-style mod
    if (ry > PI_F) ry -= TWO_PI_F;

    // size head (cols 73..75): pred * anchor + anchor
    float hh = p[73] * a0 + a0;
    float ww = p[74] * a1 + a1;
    float ll = p[75] * a2 + a2;

    // rotate by -roi_ry:  c = cos(roi_ry), s = -sin(roi_ry)
    float roi_ry = r[6];
    float sn, cs;
    sincosf(roi_ry, &sn, &cs);
    const float c =  cs;
    const float s = -sn;
    float x_rot = pos_x * c - pos_z * s;
    float z_rot = pos_x * s + pos_z * c;

    float* o = &sm_out[tid * 7];
    o[0] = x_rot + r[0];
    o[1] = pos_y;
    o[2] = z_rot + r[2];
    o[3] = hh;
    o[4] = ww;
    o[5] = ll;
    o[6] = ry + roi_ry;
  }

  __syncthreads();

  // ---- coalesced write-out of rows*7 floats ----
  {
    float* gbase = out + (size_t)base * 7;
    if (rows == ROWS_PB) {
#pragma unroll
      for (int i = 0; i < NROI_PB / THREADS; ++i) {
        const int idx = i * THREADS + tid;
        gbase[idx] = sm_out[idx];
      }
    } else {
      const int nsc = rows * 7;
#pragma unroll
      for (int i = 0; i < NROI_PB / THREADS; ++i) {
        const int idx = i * THREADS + tid;
        if (idx < nsc) gbase[idx] = sm_out[idx];
      }
    }
  }
}

extern "C" void kernel_launch(void* const* d_in, const int* in_sizes, int n_in,
                              void* d_out, int out_size, void* d_ws, size_t ws_size,
                              hipStream_t stream) {
  (void)n_in; (void)out_size; (void)d_ws; (void)ws_size;
  const float* roi    = (const float*)d_in[0];   // (N,7)
  const float* pred   = (const float*)d_in[1];   // (N,76)
  const float* anchor = (const float*)d_in[2];   // (3,)
  float* out = (float*)d_out;                    // (N,7)

  const int n = in_sizes[0] / 7;
  const int blocks = (n + ROWS_PB - 1) / ROWS_PB;
  rpn_decode_kernel<<<blocks, THREADS, 0, stream>>>(roi, pred, anchor, out, n);
}